// DGMC_24395414242144
// MI455X (gfx1250) — compile-verified
//
#include <hip/hip_runtime.h>
#include <hip/hip_bf16.h>

// ---------------------------------------------------------------------------
// DGMC forward for MI455X (gfx1250, wave32, WMMA).
// Shapes: B=8, N=512, E=65536, NT=4096, D_IN=128, D_H=64, R=32, STEPS=2.
// ---------------------------------------------------------------------------

#define NB   8
#define NN   512
#define NE   65536
#define NT   4096
#define DIN  128
#define DH   64
#define RR   32

typedef float v2f __attribute__((ext_vector_type(2)));
typedef float v8f __attribute__((ext_vector_type(8)));

__device__ __forceinline__ v8f wmma4(v2f a, v2f b, v8f c) {
  // D = A(16x4,f32) * B(4x16,f32) + C(16x16,f32)
  return __builtin_amdgcn_wmma_f32_16x16x4_f32(
      /*neg_a=*/false, a, /*neg_b=*/false, b,
      /*c_mod=*/(short)0, c, /*reuse_a=*/false, /*reuse_b=*/false);
}

// ------------------------------- utility -----------------------------------

__global__ void zero_k(float* __restrict__ p, int n) {
  int i = blockIdx.x * blockDim.x + threadIdx.x;
  if (i < n) p[i] = 0.0f;
}

// ---------------------- psi_1 scatter (128 channels) -----------------------
// agg[dst] += x[src] * ea[e]; one thread = one edge x 4 channels.
__global__ void scatter_psi1(const float* __restrict__ xs, const int* __restrict__ eis,
                             const float* __restrict__ eas,
                             const float* __restrict__ xt, const int* __restrict__ eit,
                             const float* __restrict__ eat,
                             float* __restrict__ agg_s, float* __restrict__ agg_t) {
  const int side = blockIdx.y;
  const float* x  = side ? xt  : xs;
  const int*   ei = side ? eit : eis;
  const float* ea = side ? eat : eas;
  float*       ag = side ? agg_t : agg_s;

  int idx = blockIdx.x * blockDim.x + threadIdx.x;   // E * (128/4)
  int e = idx >> 5;
  int c = (idx & 31) << 2;
  int src = ei[e];
  int dst = ei[NE + e];
  float w = ea[e];
  const float4 v = *(const float4*)(x + (size_t)src * DIN + c);
  float* p = ag + (size_t)dst * DIN + c;
  unsafeAtomicAdd(p + 0, v.x * w);
  unsafeAtomicAdd(p + 1, v.y * w);
  unsafeAtomicAdd(p + 2, v.z * w);
  unsafeAtomicAdd(p + 3, v.w * w);
}

// ---------------------- psi_2 scatter (32 channels) ------------------------
__global__ void scatter_psi2(const float* __restrict__ rs, const float* __restrict__ rt,
                             const int* __restrict__ eis, const float* __restrict__ eas,
                             const int* __restrict__ eit, const float* __restrict__ eat,
                             float* __restrict__ agg_s, float* __restrict__ agg_t) {
  const int side = blockIdx.y;
  const float* x  = side ? rt  : rs;
  const int*   ei = side ? eit : eis;
  const float* ea = side ? eat : eas;
  float*       ag = side ? agg_t : agg_s;

  int idx = blockIdx.x * blockDim.x + threadIdx.x;   // E * (32/4)
  int e = idx >> 3;
  int c = (idx & 7) << 2;
  int src = ei[e];
  int dst = ei[NE + e];
  float w = ea[e];
  const float4 v = *(const float4*)(x + (size_t)src * RR + c);
  float* p = ag + (size_t)dst * RR + c;
  unsafeAtomicAdd(p + 0, v.x * w);
  unsafeAtomicAdd(p + 1, v.y * w);
  unsafeAtomicAdd(p + 2, v.z * w);
  unsafeAtomicAdd(p + 3, v.w * w);
}

// ------------------- psi_1 dense: h = relu(xWr + aggWn + b) ----------------
// One wave (32 thr) per 16x16 output tile. K = 128 per phase, 2 phases.
__global__ void gnn1_gemm(const float* __restrict__ xs, const float* __restrict__ aggs,
                          const float* __restrict__ xt, const float* __restrict__ aggt,
                          const float* __restrict__ Wr, const float* __restrict__ Wn,
                          const float* __restrict__ bias,
                          float* __restrict__ hs, float* __restrict__ ht) {
  const int side = blockIdx.z;
  const float* X = side ? xt   : xs;
  const float* A = side ? aggt : aggs;
  float*       H = side ? ht   : hs;
  const int m0 = blockIdx.x * 16, n0 = blockIdx.y * 16;
  const int lane = threadIdx.x, hi = lane >> 4, l = lane & 15;

  v8f c = {};
  const float* xrow = X + (size_t)(m0 + l) * DIN;
  const float* arow = A + (size_t)(m0 + l) * DIN;
  for (int k = 0; k < DIN; k += 4) {
    int ka = k + 2 * hi;
    v2f a = { xrow[ka], xrow[ka + 1] };
    v2f b = { Wr[ka * DH + n0 + l], Wr[(ka + 1) * DH + n0 + l] };
    c = wmma4(a, b, c);
  }
  for (int k = 0; k < DIN; k += 4) {
    int ka = k + 2 * hi;
    v2f a = { arow[ka], arow[ka + 1] };
    v2f b = { Wn[ka * DH + n0 + l], Wn[(ka + 1) * DH + n0 + l] };
    c = wmma4(a, b, c);
  }
  float bb = bias[n0 + l];
#pragma unroll
  for (int i = 0; i < 8; ++i) {
    int row = m0 + i + 8 * hi;
    H[(size_t)row * DH + n0 + l] = fmaxf(c[i] + bb, 0.0f);
  }
}

// ------------------- S_hat0[b] = h_s[b] @ h_t[b]^T  (K=64) -----------------
__global__ void shat0_gemm(const float* __restrict__ hs, const float* __restrict__ ht,
                           float* __restrict__ shat) {
  const int b = blockIdx.z;
  const int m0 = blockIdx.x * 16, n0 = blockIdx.y * 16;
  const int lane = threadIdx.x, hi = lane >> 4, l = lane & 15;

  const float* As = hs + (size_t)b * NN * DH + (size_t)(m0 + l) * DH;
  const float* Bt = ht + (size_t)b * NN * DH + (size_t)(n0 + l) * DH;
  v8f c = {};
  for (int k = 0; k < DH; k += 4) {
    int ka = k + 2 * hi;
    v2f a  = { As[ka], As[ka + 1] };
    v2f bb = { Bt[ka], Bt[ka + 1] };   // B[k][n] = h_t[n][k]
    c = wmma4(a, bb, c);
  }
  float* out = shat + (size_t)b * NN * NN;
#pragma unroll
  for (int i = 0; i < 8; ++i)
    out[(size_t)(m0 + i + 8 * hi) * NN + n0 + l] = c[i];
}

// ----------------- r_t[b] = S[b]^T @ r_s[b]   ([512,512]^T x [512,32]) -----
__global__ void rt_gemm(const float* __restrict__ S, const float* __restrict__ rs,
                        float* __restrict__ rt) {
  const int b = blockIdx.z;
  const int m0 = blockIdx.x * 16, n0 = blockIdx.y * 16;
  const int lane = threadIdx.x, hi = lane >> 4, l = lane & 15;

  const float* Sb = S  + (size_t)b * NN * NN;
  const float* rb = rs + (size_t)b * NN * RR;
  v8f c = {};
  for (int k = 0; k < NN; k += 4) {
    int ka = k + 2 * hi;
    v2f a  = { Sb[(size_t)ka * NN + m0 + l], Sb[(size_t)(ka + 1) * NN + m0 + l] };
    v2f bb = { rb[(size_t)ka * RR + n0 + l], rb[(size_t)(ka + 1) * RR + n0 + l] };
    c = wmma4(a, bb, c);
  }
  float* out = rt + (size_t)b * NN * RR;
#pragma unroll
  for (int i = 0; i < 8; ++i)
    out[(size_t)(m0 + i + 8 * hi) * RR + n0 + l] = c[i];
}

// ------------- psi_2 dense: o = relu(r W2r + agg W2n + b2)  (K=32) ---------
__global__ void psi2_gemm(const float* __restrict__ rs, const float* __restrict__ rt,
                          const float* __restrict__ aggs, const float* __restrict__ aggt,
                          const float* __restrict__ Wr, const float* __restrict__ Wn,
                          const float* __restrict__ bias,
                          float* __restrict__ os, float* __restrict__ ot) {
  const int side = blockIdx.z;
  const float* X = side ? rt   : rs;
  const float* A = side ? aggt : aggs;
  float*       O = side ? ot   : os;
  const int m0 = blockIdx.x * 16, n0 = blockIdx.y * 16;
  const int lane = threadIdx.x, hi = lane >> 4, l = lane & 15;

  v8f c = {};
  const float* xrow = X + (size_t)(m0 + l) * RR;
  const float* arow = A + (size_t)(m0 + l) * RR;
  for (int k = 0; k < RR; k += 4) {
    int ka = k + 2 * hi;
    v2f a = { xrow[ka], xrow[ka + 1] };
    v2f b = { Wr[ka * RR + n0 + l], Wr[(ka + 1) * RR + n0 + l] };
    c = wmma4(a, b, c);
  }
  for (int k = 0; k < RR; k += 4) {
    int ka = k + 2 * hi;
    v2f a = { arow[ka], arow[ka + 1] };
    v2f b = { Wn[ka * RR + n0 + l], Wn[(ka + 1) * RR + n0 + l] };
    c = wmma4(a, b, c);
  }
  float bb = bias[n0 + l];
#pragma unroll
  for (int i = 0; i < 8; ++i)
    O[(size_t)(m0 + i + 8 * hi) * RR + n0 + l] = fmaxf(c[i] + bb, 0.0f);
}

// ------------- a = o @ Wm1 (+ bm1 on s-side only)  (K=32) ------------------
__global__ void aproj_gemm(const float* __restrict__ os, const float* __restrict__ ot,
                           const float* __restrict__ Wm1, const float* __restrict__ bm1,
                           float* __restrict__ as_, float* __restrict__ at_) {
  const int side = blockIdx.z;
  const float* O   = side ? ot  : os;
  float*       OUT = side ? at_ : as_;
  const int m0 = blockIdx.x * 16, n0 = blockIdx.y * 16;
  const int lane = threadIdx.x, hi = lane >> 4, l = lane & 15;

  v8f c = {};
  const float* orow = O + (size_t)(m0 + l) * RR;
  for (int k = 0; k < RR; k += 4) {
    int ka = k + 2 * hi;
    v2f a = { orow[ka], orow[ka + 1] };
    v2f b = { Wm1[ka * RR + n0 + l], Wm1[(ka + 1) * RR + n0 + l] };
    c = wmma4(a, b, c);
  }
  float bb = side ? 0.0f : bm1[n0 + l];
#pragma unroll
  for (int i = 0; i < 8; ++i)
    OUT[(size_t)(m0 + i + 8 * hi) * RR + n0 + l] = c[i] + bb;
}

// ------ S_hat[b,s,t] += bm2 + sum_r relu(a_s[b,s,r]-a_t[b,t,r])*Wm2[r] -----
__global__ void consensus_k(const float* __restrict__ as_, const float* __restrict__ at_,
                            const float* __restrict__ Wm2, const float* __restrict__ bm2,
                            float* __restrict__ shat) {
  const int bs = blockIdx.x;          // b*512 + s
  const int b = bs >> 9;
  __shared__ float arow[RR];
  __shared__ float w2[RR];
  const int t = threadIdx.x;
  if (t < RR) { arow[t] = as_[(size_t)bs * RR + t]; w2[t] = Wm2[t]; }
  __syncthreads();
  const float bb = bm2[0];
  const float* atb = at_ + (size_t)b * NN * RR;
  float* srow = shat + (size_t)bs * NN;
  for (int tt = t; tt < NN; tt += blockDim.x) {
    const float* ar = atb + (size_t)tt * RR;
    float acc = bb;
#pragma unroll
    for (int r = 0; r < RR; ++r)
      acc += fmaxf(arow[r] - ar[r], 0.0f) * w2[r];
    srow[tt] += acc;
  }
}

// ------------------------ row softmax over 512 -----------------------------
__global__ void softmax_rows(const float* __restrict__ in, float* __restrict__ out) {
  const int row = blockIdx.x;         // B*N rows
  const float* p = in  + (size_t)row * NN;
  float*       q = out + (size_t)row * NN;
  const int t = threadIdx.x;          // 256 threads, 2 elems each
  float v0 = p[t], v1 = p[t + 256];
  __shared__ float red[256];
  red[t] = fmaxf(v0, v1);
  __syncthreads();
  for (int s = 128; s > 0; s >>= 1) {
    if (t < s) red[t] = fmaxf(red[t], red[t + s]);
    __syncthreads();
  }
  const float m = red[0];
  __syncthreads();
  float e0 = __expf(v0 - m), e1 = __expf(v1 - m);
  red[t] = e0 + e1;
  __syncthreads();
  for (int s = 128; s > 0; s >>= 1) {
    if (t < s) red[t] += red[t + s];
    __syncthreads();
  }
  const float inv = 1.0f / red[0];
  q[t] = e0 * inv;
  q[t + 256] = e1 * inv;
}

// ---------------------------------------------------------------------------

extern "C" void kernel_launch(void* const* d_in, const int* in_sizes, int n_in,
                              void* d_out, int out_size, void* d_ws, size_t ws_size,
                              hipStream_t stream) {
  const float* x_s  = (const float*)d_in[0];
  const int*   ei_s = (const int*)  d_in[1];
  const float* ea_s = (const float*)d_in[2];
  const float* x_t  = (const float*)d_in[4];
  const int*   ei_t = (const int*)  d_in[5];
  const float* ea_t = (const float*)d_in[6];
  const float* r_all= (const float*)d_in[8];
  const float* W1r  = (const float*)d_in[9];
  const float* W1n  = (const float*)d_in[10];
  const float* b1   = (const float*)d_in[11];
  const float* W2r  = (const float*)d_in[12];
  const float* W2n  = (const float*)d_in[13];
  const float* b2   = (const float*)d_in[14];
  const float* Wm1  = (const float*)d_in[15];
  const float* bm1  = (const float*)d_in[16];
  const float* Wm2  = (const float*)d_in[17];
  const float* bm2  = (const float*)d_in[18];

  float* ws = (float*)d_ws;
  float* agg1_s = ws;                       // NT*DIN
  float* agg1_t = agg1_s + NT * DIN;        // NT*DIN (contiguous with agg1_s)
  float* h_s    = agg1_t + NT * DIN;        // NT*DH
  float* h_t    = h_s    + NT * DH;
  float* S_hat  = h_t    + NT * DH;         // B*N*N
  float* S_scr  = S_hat  + NB * NN * NN;    // B*N*N
  float* r_t    = S_scr  + NB * NN * NN;    // NT*RR
  float* agg2_s = r_t    + NT * RR;         // NT*RR
  float* agg2_t = agg2_s + NT * RR;         // NT*RR (contiguous with agg2_s)
  float* o_s    = agg2_t + NT * RR;
  float* o_t    = o_s    + NT * RR;
  float* a_s    = o_t    + NT * RR;
  float* a_t    = a_s    + NT * RR;

  float* S0_out = (float*)d_out;
  float* SL_out = S0_out + (size_t)NB * NN * NN;

  // ---- psi_1 ----
  {
    const int n = 2 * NT * DIN;
    zero_k<<<(n + 255) / 256, 256, 0, stream>>>(agg1_s, n);
  }
  scatter_psi1<<<dim3(NE * 32 / 256, 2), 256, 0, stream>>>(
      x_s, ei_s, ea_s, x_t, ei_t, ea_t, agg1_s, agg1_t);
  gnn1_gemm<<<dim3(NT / 16, DH / 16, 2), 32, 0, stream>>>(
      x_s, agg1_s, x_t, agg1_t, W1r, W1n, b1, h_s, h_t);

  // ---- S_hat0 and S_0 ----
  shat0_gemm<<<dim3(NN / 16, NN / 16, NB), 32, 0, stream>>>(h_s, h_t, S_hat);
  softmax_rows<<<NB * NN, 256, 0, stream>>>(S_hat, S0_out);

  // ---- consensus steps ----
  for (int step = 0; step < 2; ++step) {
    const float* rs = r_all + (size_t)step * NB * NN * RR;
    const float* S;
    if (step == 0) {
      S = S0_out;                              // softmax(S_hat0) already computed
    } else {
      softmax_rows<<<NB * NN, 256, 0, stream>>>(S_hat, S_scr);
      S = S_scr;
    }
    rt_gemm<<<dim3(NN / 16, RR / 16, NB), 32, 0, stream>>>(S, rs, r_t);

    {
      const int n = 2 * NT * RR;
      zero_k<<<(n + 255) / 256, 256, 0, stream>>>(agg2_s, n);
    }
    scatter_psi2<<<dim3(NE * 8 / 256, 2), 256, 0, stream>>>(
        rs, r_t, ei_s, ea_s, ei_t, ea_t, agg2_s, agg2_t);
    psi2_gemm<<<dim3(NT / 16, RR / 16, 2), 32, 0, stream>>>(
        rs, r_t, agg2_s, agg2_t, W2r, W2n, b2, o_s, o_t);
    aproj_gemm<<<dim3(NT / 16, RR / 16, 2), 32, 0, stream>>>(
        o_s, o_t, Wm1, bm1, a_s, a_t);
    consensus_k<<<NB * NN, 256, 0, stream>>>(a_s, a_t, Wm2, bm2, S_hat);
  }

  // ---- S_L ----
  softmax_rows<<<NB * NN, 256, 0, stream>>>(S_hat, SL_out);
}